// DecoderLSTM_5394478923820
// MI455X (gfx1250) — compile-verified
//
#include <hip/hip_runtime.h>
#include <math.h>

// ---------------------------------------------------------------------------
// DecoderLSTM single step for MI455X (gfx1250, wave32).
// Memory-bound GEMV problem (~237 MB fp32 weights, ~10us @ 23.3 TB/s floor).
// GEMVs are executed with V_WMMA_F32_16X16X4_F32: A = 16x4 weight tile,
// B = x-chunk broadcast to all 16 columns, accumulate in C over K.
// ---------------------------------------------------------------------------

typedef __attribute__((ext_vector_type(2))) float v2f;
typedef __attribute__((ext_vector_type(8))) float v8f;

#define HDIM 1024
#define G4   4096
#define VOCAB 50257

// workspace layout (float offsets)
#define WS_GATES 0      // 4096 floats
#define WS_H1    4096   // 1024 floats
#define WS_RED   5120   // 1 float (logZ)

// Accumulate 16 rows of W (row indices rowLane = per-lane, pre-clamped) dotted
// with xs[0..H) (LDS) into acc via chained f32 WMMA.  Wave-uniform control
// flow; EXEC must be all 1s on entry.
__device__ __forceinline__ v8f wmma_gemv_rows16(const float* __restrict__ W,
                                                const float* __restrict__ xs,
                                                int rowLane, v8f acc)
{
  const int lane = threadIdx.x & 31;
  const int kh4  = (lane >> 4) << 2;                 // 0 for lanes 0-15, 4 for 16-31
  const float* __restrict__ wrow = W + (size_t)rowLane * HDIM;
#if __has_builtin(__builtin_amdgcn_wmma_f32_16x16x4_f32)
  #pragma unroll 4
  for (int k0 = 0; k0 < HDIM; k0 += 8) {
    const float4 w  = *(const float4*)(wrow + k0 + kh4);   // global_load_b128
    const float4 xv = *(const float4*)(xs  + k0 + kh4);    // ds_load_b128
    v2f a, b;
    // WMMA #1: logical K 0..3 <-> physical cols {k0,k0+1,k0+4,k0+5}
    a[0] = w.x;  a[1] = w.y;
    b[0] = xv.x; b[1] = xv.y;
    acc = __builtin_amdgcn_wmma_f32_16x16x4_f32(false, a, false, b,
                                                (short)0, acc, false, false);
    // WMMA #2: physical cols {k0+2,k0+3,k0+6,k0+7}
    a[0] = w.z;  a[1] = w.w;
    b[0] = xv.z; b[1] = xv.w;
    acc = __builtin_amdgcn_wmma_f32_16x16x4_f32(false, a, false, b,
                                                (short)0, acc, false, false);
  }
#else
  // fallback: plain FMA dot (should not be taken on gfx1250 toolchains)
  const int m = lane & 15;
  (void)m;
  float s = 0.f;
  for (int k0 = 0; k0 < HDIM; k0 += 8) {
    const float4 w  = *(const float4*)(wrow + k0 + kh4);
    const float4 xv = *(const float4*)(xs  + k0 + kh4);
    s += w.x * xv.x + w.y * xv.y + w.z * xv.z + w.w * xv.w;
  }
  acc[0] += s;
#endif
  return acc;
}

// ---- kernel 1: gates = W_ih * relu(emb[tok]) + W_hh * h0 + b_ih + b_hh -----
__global__ __launch_bounds__(256)
void lstm_gates_kernel(const int* __restrict__ token,
                       const float* __restrict__ emb,
                       const float* __restrict__ h0,
                       const float* __restrict__ W_ih,
                       const float* __restrict__ W_hh,
                       const float* __restrict__ b_ih,
                       const float* __restrict__ b_hh,
                       float* __restrict__ ws)
{
  __shared__ __align__(16) float xs[HDIM];
  __shared__ __align__(16) float hs[HDIM];
  const int tid = threadIdx.x;
  const int tok = token[0];
  for (int i = tid; i < HDIM; i += 256) {
    const float e = emb[(size_t)tok * HDIM + i];
    xs[i] = e > 0.f ? e : 0.f;                       // relu(embedding)
    hs[i] = h0[i];
  }
  __syncthreads();

  const int wave = blockIdx.x * 8 + (tid >> 5);      // 256 waves total
  const int row0 = wave << 4;                        // 16 rows per wave
  const int lane = tid & 31;
  const int rowLane = row0 + (lane & 15);

  v8f acc = {};
  acc = wmma_gemv_rows16(W_ih, xs, rowLane, acc);
  acc = wmma_gemv_rows16(W_hh, hs, rowLane, acc);

  if (lane == 0 || lane == 16) {                     // column 0 of D holds y
    const int rbase = row0 + ((lane >> 4) << 3);
    #pragma unroll
    for (int r = 0; r < 8; ++r) {
      const int row = rbase + r;
      ws[WS_GATES + row] = acc[r] + b_ih[row] + b_hh[row];
    }
  }
}

// ---- kernel 2: LSTM pointwise cell; writes h1/c1 into d_out slots ----------
__global__ __launch_bounds__(256)
void lstm_cell_kernel(const float* __restrict__ c0,
                      float* __restrict__ ws,
                      float* __restrict__ out)
{
  const int i = blockIdx.x * 256 + threadIdx.x;
  if (i >= HDIM) return;
  const float gi = ws[WS_GATES + i];
  const float gf = ws[WS_GATES + HDIM + i];
  const float gg = ws[WS_GATES + 2 * HDIM + i];
  const float go = ws[WS_GATES + 3 * HDIM + i];
  const float si = 1.f / (1.f + expf(-gi));
  const float sf = 1.f / (1.f + expf(-gf));
  const float so = 1.f / (1.f + expf(-go));
  const float c1 = sf * c0[i] + si * tanhf(gg);
  const float h1 = so * tanhf(c1);
  ws[WS_H1 + i]        = h1;
  out[VOCAB + i]        = h1;   // output slot 1: hidden
  out[VOCAB + HDIM + i] = c1;   // output slot 2: cell
}

// ---- kernel 3: logits = W_out * h1 + b_out  -> d_out[0..V) -----------------
__global__ __launch_bounds__(256)
void out_gemv_kernel(const float* __restrict__ W_out,
                     const float* __restrict__ b_out,
                     const float* __restrict__ ws,
                     float* __restrict__ out)
{
  __shared__ __align__(16) float hs[HDIM];
  const int tid = threadIdx.x;
  for (int i = tid; i < HDIM; i += 256) hs[i] = ws[WS_H1 + i];
  __syncthreads();

  const int wave = blockIdx.x * 8 + (tid >> 5);
  const int row0 = wave << 4;
  if (row0 >= VOCAB) return;                          // wave-uniform exit

  const int lane = tid & 31;
  int rowLane = row0 + (lane & 15);
  if (rowLane > VOCAB - 1) rowLane = VOCAB - 1;       // clamp tail loads

  v8f acc = {};
  acc = wmma_gemv_rows16(W_out, hs, rowLane, acc);

  if (lane == 0 || lane == 16) {
    const int rbase = row0 + ((lane >> 4) << 3);
    #pragma unroll
    for (int r = 0; r < 8; ++r) {
      const int row = rbase + r;
      if (row < VOCAB) out[row] = acc[r] + b_out[row];
    }
  }
}

// ---- kernel 4: single-block max + log-sum-exp over logits ------------------
__global__ __launch_bounds__(1024)
void softmax_reduce_kernel(const float* __restrict__ out,
                           float* __restrict__ ws)
{
  __shared__ float red[1024];
  const int tid = threadIdx.x;
  float m = -INFINITY;
  for (int i = tid; i < VOCAB; i += 1024) m = fmaxf(m, out[i]);
  red[tid] = m;
  __syncthreads();
  for (int s = 512; s > 0; s >>= 1) {
    if (tid < s) red[tid] = fmaxf(red[tid], red[tid + s]);
    __syncthreads();
  }
  const float gmax = red[0];
  __syncthreads();
  float sum = 0.f;
  for (int i = tid; i < VOCAB; i += 1024) sum += expf(out[i] - gmax);
  red[tid] = sum;
  __syncthreads();
  for (int s = 512; s > 0; s >>= 1) {
    if (tid < s) red[tid] += red[tid + s];
    __syncthreads();
  }
  if (tid == 0) ws[WS_RED] = gmax + logf(red[0]);
}

// ---- kernel 5: in-place log-softmax ----------------------------------------
__global__ __launch_bounds__(256)
void logsoftmax_kernel(float* __restrict__ out, const float* __restrict__ ws)
{
  const int i = blockIdx.x * 256 + threadIdx.x;
  if (i < VOCAB) out[i] -= ws[WS_RED];
}

// ---------------------------------------------------------------------------
extern "C" void kernel_launch(void* const* d_in, const int* in_sizes, int n_in,
                              void* d_out, int out_size, void* d_ws, size_t ws_size,
                              hipStream_t stream)
{
  (void)in_sizes; (void)n_in; (void)out_size; (void)ws_size;
  const int*   token = (const int*)  d_in[0];
  const float* h0    = (const float*)d_in[1];
  const float* c0    = (const float*)d_in[2];
  const float* emb   = (const float*)d_in[3];
  const float* W_ih  = (const float*)d_in[4];
  const float* W_hh  = (const float*)d_in[5];
  const float* b_ih  = (const float*)d_in[6];
  const float* b_hh  = (const float*)d_in[7];
  const float* W_out = (const float*)d_in[8];
  const float* b_out = (const float*)d_in[9];
  float* out = (float*)d_out;
  float* ws  = (float*)d_ws;

  // 4096 gate rows / 16 rows-per-wave / 8 waves-per-block = 32 blocks
  lstm_gates_kernel<<<32, 256, 0, stream>>>(token, emb, h0, W_ih, W_hh,
                                            b_ih, b_hh, ws);
  lstm_cell_kernel<<<(HDIM + 255) / 256, 256, 0, stream>>>(c0, ws, out);

  const int waves  = (VOCAB + 15) / 16;   // 3142
  const int blocks = (waves + 7) / 8;     // 393
  out_gemv_kernel<<<blocks, 256, 0, stream>>>(W_out, b_out, ws, out);

  softmax_reduce_kernel<<<1, 1024, 0, stream>>>(out, ws);
  logsoftmax_kernel<<<(VOCAB + 255) / 256, 256, 0, stream>>>(out, ws);
}